// SOM_12146167513220
// MI455X (gfx1250) — compile-verified
//
#include <hip/hip_runtime.h>

#define HGRID 100
#define WGRID 100
#define HW    10000
#define DIN   256
#define BATCH 1024

typedef __attribute__((ext_vector_type(2))) float v2f;
typedef __attribute__((ext_vector_type(8))) float v8f;

__device__ __forceinline__ unsigned long long shfl_xor_u64(unsigned long long v, int mask, int width) {
    unsigned int lo = (unsigned int)v;
    unsigned int hi = (unsigned int)(v >> 32);
    lo = __shfl_xor(lo, mask, width);
    hi = __shfl_xor(hi, mask, width);
    return ((unsigned long long)hi << 32) | lo;
}

// ---- transpose x (BATCH x DIN) -> xT (DIN x BATCH), staged through LDS via
// ---- CDNA5 async global->LDS loads (ASYNCcnt path). One 16x16 tile / block.
__global__ void k_transpose_x(const float* __restrict__ x, float* __restrict__ xT) {
    __shared__ float tile[16 * 17]; // padded: no bank conflicts on transposed read
    int t  = threadIdx.x;
    int tx = t & 15, ty = t >> 4;
    int b0 = blockIdx.x * 16, d0 = blockIdx.y * 16;

    const float* gsrc = x + (size_t)(b0 + ty) * DIN + d0 + tx;
    unsigned lds_off = (unsigned)((ty * 17 + tx) * 4); // 'tile' is the only LDS object -> base 0
    asm volatile("global_load_async_to_lds_b32 %0, %1, off"
                 :: "v"(lds_off), "v"(gsrc) : "memory");
    asm volatile("s_wait_asynccnt 0" ::: "memory");
    __syncthreads();

    xT[(size_t)(d0 + ty) * BATCH + b0 + tx] = tile[tx * 17 + ty];
}

// ---- |w|^2 per weight row (8 rows per 256-thread block) ----
__global__ void k_wnorm(const float* __restrict__ w, float* __restrict__ wn2) {
    int wave = threadIdx.x >> 5;
    int lane = threadIdx.x & 31;
    int row  = blockIdx.x * 8 + wave; // exact: 1250*8 = HW
    const float* wr = w + (size_t)row * DIN;
    float s = 0.f;
#pragma unroll
    for (int j = 0; j < 8; ++j) {
        float v = wr[lane + 32 * j];
        s += v * v;
    }
#pragma unroll
    for (int off = 16; off; off >>= 1) s += __shfl_down(s, off, 32);
    if (lane == 0) wn2[row] = s;
}

__global__ void k_init_bmu(unsigned long long* __restrict__ bmu) {
    int i = blockIdx.x * blockDim.x + threadIdx.x;
    if (i < BATCH) bmu[i] = ~0ull;
}

// ---- GEMM1 (f32 WMMA 16x16x4): dot = x @ W^T tile, fused argmin of
// ---- (|w|^2 - 2 x.w) via packed-u64 lane reduce + atomicMin. 1 wave / 16x16 tile.
__global__ void k_gemm1_argmin(const float* __restrict__ x,
                               const float* __restrict__ w,
                               const float* __restrict__ wn2,
                               unsigned long long* __restrict__ bmu) {
    int lane = threadIdx.x;
    int l16  = lane & 15;
    int half = lane >> 4;
    int bBase  = blockIdx.x * 16;
    int hwBase = blockIdx.y * 16;

    const float* aRow = x + (size_t)(bBase + l16) * DIN + half * 2;  // A row = batch
    const float* bCol = w + (size_t)(hwBase + l16) * DIN + half * 2; // B col = W row

    v8f acc = {};
#pragma unroll 8
    for (int kc = 0; kc < 64; ++kc) { // K = 256 in steps of 4
        if ((kc & 7) == 0) {
            __builtin_prefetch(aRow + kc * 4 + 32, 0, 1);
            __builtin_prefetch(bCol + kc * 4 + 32, 0, 1);
        }
        v2f a = *(const v2f*)(aRow + kc * 4);
        v2f b = *(const v2f*)(bCol + kc * 4);
        acc = __builtin_amdgcn_wmma_f32_16x16x4_f32(false, a, false, b,
                                                    (short)0, acc, false, false);
    }

    float wn = wn2[hwBase + l16];
    unsigned int hwIdx = (unsigned int)(hwBase + l16);
#pragma unroll
    for (int r = 0; r < 8; ++r) {
        float v = wn - 2.0f * acc[r];
        unsigned int u = __float_as_uint(v);
        u = (u & 0x80000000u) ? ~u : (u | 0x80000000u); // order-preserving map
        unsigned long long key = ((unsigned long long)u << 32) | hwIdx;
#pragma unroll
        for (int off = 8; off; off >>= 1) {
            unsigned long long o = shfl_xor_u64(key, off, 16);
            if (o < key) key = o;
        }
        if (l16 == 0) atomicMin(bmu + (bBase + r + half * 8), key);
    }
}

// ---- decode BMU index -> (by, bx) floats ----
__global__ void k_decode(const unsigned long long* __restrict__ bmu, float* __restrict__ byx) {
    int i = blockIdx.x * blockDim.x + threadIdx.x;
    if (i < BATCH) {
        unsigned int idx = (unsigned int)(bmu[i] & 0xFFFFFFFFu);
        byx[i]         = (float)(idx / WGRID);
        byx[BATCH + i] = (float)(idx % WGRID);
    }
}

// ---- suffix-product coefficients: Ct (HW x BATCH) f32 row-major, A0 (HW) ----
// c_i(hw) = a_i * prod_{j>i} (1 - a_j),  a_i = lr * exp(-grid_d2 / (2 r^2))
__global__ void k_coeff(const float* __restrict__ byx,
                        const int* __restrict__ epochp, const int* __restrict__ totalp,
                        float* __restrict__ Ct, float* __restrict__ A0) {
    int hw = blockIdx.x * blockDim.x + threadIdx.x;
    if (hw >= HW) return;
    float h = (float)(hw / WGRID);
    float w = (float)(hw % WGRID);
    float decay  = 1.0f - (float)epochp[0] / (float)totalp[0];
    float lr     = 0.5f  * decay;
    float radius = 50.0f * decay;
    float inv2r2 = 1.0f / (2.0f * radius * radius);
    float S = 1.0f;
    float4 buf;
    float* row = Ct + (size_t)hw * BATCH;
    for (int i = BATCH - 1; i >= 0; --i) {
        float dy = byx[i] - h;
        float dx = byx[BATCH + i] - w;
        float n  = __expf(-(dy * dy + dx * dx) * inv2r2);
        float a  = lr * n;
        ((float*)&buf)[i & 3] = a * S;
        if ((i & 3) == 0) *(float4*)(row + i) = buf; // 16B stores
        S *= (1.0f - a);
    }
    A0[hw] = S;
}

// ---- GEMM2 (f32 WMMA 16x16x4): out = A0 .* w_old + C @ X ----
// A = Ct (HW x BATCH, rows contiguous), B = xT (DIN x BATCH, cols contiguous)
__global__ void k_gemm2(const float* __restrict__ Ct,
                        const float* __restrict__ xT,
                        const float* __restrict__ A0,
                        const float* __restrict__ wold,
                        float* __restrict__ out) {
    int lane = threadIdx.x;
    int l16  = lane & 15;
    int half = lane >> 4;
    int hwBase = blockIdx.x * 16;
    int dBase  = blockIdx.y * 16;

    const float* aRow = Ct + (size_t)(hwBase + l16) * BATCH + half * 2; // A row = hw
    const float* bCol = xT + (size_t)(dBase + l16) * BATCH + half * 2;  // B col = d

    v8f acc = {};
#pragma unroll 8
    for (int kc = 0; kc < 256; ++kc) { // K = BATCH in steps of 4
        if ((kc & 7) == 0) {
            __builtin_prefetch(aRow + kc * 4 + 32, 0, 1);
            __builtin_prefetch(bCol + kc * 4 + 32, 0, 1);
        }
        v2f a = *(const v2f*)(aRow + kc * 4);
        v2f b = *(const v2f*)(bCol + kc * 4);
        acc = __builtin_amdgcn_wmma_f32_16x16x4_f32(false, a, false, b,
                                                    (short)0, acc, false, false);
    }

#pragma unroll
    for (int r = 0; r < 8; ++r) {
        int hw = hwBase + r + half * 8;
        int d  = dBase + l16;
        out[(size_t)hw * DIN + d] = A0[hw] * wold[(size_t)hw * DIN + d] + acc[r];
    }
}

extern "C" void kernel_launch(void* const* d_in, const int* in_sizes, int n_in,
                              void* d_out, int out_size, void* d_ws, size_t ws_size,
                              hipStream_t stream) {
    (void)in_sizes; (void)n_in; (void)out_size; (void)ws_size;
    const float* x  = (const float*)d_in[0];
    const float* w  = (const float*)d_in[1];
    const int*   ep = (const int*)d_in[2];
    const int*   te = (const int*)d_in[3];
    float* out = (float*)d_out;

    char* ws = (char*)d_ws;
    size_t off = 0;
    auto take = [&](size_t bytes) -> char* {
        char* p = ws + off;
        off = (off + bytes + 255) & ~(size_t)255;
        return p;
    };
    float*              xT  = (float*)take((size_t)DIN * BATCH * 4);
    float*              wn2 = (float*)take((size_t)HW * 4);
    unsigned long long* bmu = (unsigned long long*)take((size_t)BATCH * 8);
    float*              byx = (float*)take((size_t)BATCH * 2 * 4);
    float*              Ct  = (float*)take((size_t)HW * BATCH * 4);
    float*              A0  = (float*)take((size_t)HW * 4);

    k_transpose_x <<<dim3(BATCH / 16, DIN / 16), dim3(256), 0, stream>>>(x, xT);
    k_wnorm       <<<dim3(HW / 8),               dim3(256), 0, stream>>>(w, wn2);
    k_init_bmu    <<<dim3(4),                    dim3(256), 0, stream>>>(bmu);
    k_gemm1_argmin<<<dim3(BATCH / 16, HW / 16),  dim3(32),  0, stream>>>(x, w, wn2, bmu);
    k_decode      <<<dim3(4),                    dim3(256), 0, stream>>>(bmu, byx);
    k_coeff       <<<dim3(40),                   dim3(256), 0, stream>>>(byx, ep, te, Ct, A0);
    k_gemm2       <<<dim3(HW / 16, DIN / 16),    dim3(32),  0, stream>>>(Ct, xT, A0, w, out);
}